// PointPatchify_72241349919076
// MI455X (gfx1250) — compile-verified
//
#include <hip/hip_runtime.h>
#include <float.h>

#define NPTS   65536
#define NBATCH 16
#define NPATCH 64
#define PSIZE  32
#define ROWP   2052   // 2048 + 4 pad: float4-aligned rows, conflict-free WMMA stores

typedef __attribute__((ext_vector_type(2))) float v2f;
typedef __attribute__((ext_vector_type(8))) float v8f;

// ---------------------------------------------------------------------------
// FPS: one block per batch. x,y + min-dist in VGPRs, z in LDS (256KB).
// Each thread owns 16 quads of 4 consecutive points -> float4 LDS/global I/O.
// 63 rounds of update + block argmax (ties -> lowest index, like jnp.argmax).
// ---------------------------------------------------------------------------
__global__ __launch_bounds__(1024) void fps_kernel(const float* __restrict__ points,
                                                   float* __restrict__ out)
{
    extern __shared__ float zb[];          // NPTS floats = 256 KB
    __shared__ float rv[32];
    __shared__ int   ri[32];
    __shared__ float sLx, sLy, sLz;

    const int tid = threadIdx.x;
    const int b   = blockIdx.x;
    const float* pts = points + (size_t)b * NPTS * 3;
    float* ctr = out + (size_t)NBATCH * NPATCH * PSIZE * 3 + (size_t)b * NPATCH * 3;

    float px[64], py[64], md[64];
    {
        const float4* p4  = (const float4*)pts;
        float4*       zb4 = (float4*)zb;
        #pragma unroll
        for (int q = 0; q < 16; ++q) {
            const int n0 = (tid << 2) + (q << 12);      // first point of quad
            const int f4 = 3 * tid + (q << 10) * 3;     // n0*3/4
            const float4 a = p4[f4 + 0];
            const float4 c = p4[f4 + 1];
            const float4 e = p4[f4 + 2];
            px[q*4+0] = a.x; py[q*4+0] = a.y;
            px[q*4+1] = a.w; py[q*4+1] = c.x;
            px[q*4+2] = c.z; py[q*4+2] = c.w;
            px[q*4+3] = e.y; py[q*4+3] = e.z;
            zb4[n0 >> 2] = make_float4(a.z, c.y, e.x, e.w);
            md[q*4+0] = FLT_MAX; md[q*4+1] = FLT_MAX;
            md[q*4+2] = FLT_MAX; md[q*4+3] = FLT_MAX;
        }
    }
    if (tid == 0) { sLx = pts[0]; sLy = pts[1]; sLz = pts[2]; }
    __syncthreads();

    const int lane = tid & 31, wv = tid >> 5;
    const float4* zb4 = (const float4*)zb;

    for (int k = 0; k < NPATCH; ++k) {
        if (tid == 0) { ctr[k*3+0] = sLx; ctr[k*3+1] = sLy; ctr[k*3+2] = sLz; }
        if (k == NPATCH - 1) break;

        const float lx = sLx, ly = sLy, lz = sLz;
        float bestV = -FLT_MAX; int bestI = 0x7fffffff;
        #pragma unroll
        for (int q = 0; q < 16; ++q) {
            const float4 zq = zb4[tid + (q << 10)];
            #pragma unroll
            for (int r = 0; r < 4; ++r) {
                const int i = q*4 + r;
                const int n = (tid << 2) + r + (q << 12);
                const float zr = (r==0) ? zq.x : (r==1) ? zq.y : (r==2) ? zq.z : zq.w;
                const float dx = px[i] - lx, dy = py[i] - ly, dz = zr - lz;
                const float d = fmaf(dz, dz, fmaf(dy, dy, dx * dx));
                const float m = fminf(md[i], d);
                md[i] = m;
                if (m > bestV) { bestV = m; bestI = n; }  // strict > keeps lowest index
            }
        }
        // wave32 butterfly argmax, ties -> lower index
        #pragma unroll
        for (int off = 16; off > 0; off >>= 1) {
            const float ov = __shfl_xor(bestV, off);
            const int   oi = __shfl_xor(bestI, off);
            if (ov > bestV || (ov == bestV && oi < bestI)) { bestV = ov; bestI = oi; }
        }
        if (lane == 0) { rv[wv] = bestV; ri[wv] = bestI; }
        __syncthreads();
        if (wv == 0) {
            float v = rv[lane]; int idx = ri[lane];
            #pragma unroll
            for (int off = 16; off > 0; off >>= 1) {
                const float ov = __shfl_xor(v, off);
                const int   oi = __shfl_xor(idx, off);
                if (ov > v || (ov == v && oi < idx)) { v = ov; idx = oi; }
            }
            if (lane == 0) {
                sLx = pts[idx * 3 + 0];
                sLy = pts[idx * 3 + 1];
                sLz = zb[idx];
            }
        }
        __syncthreads();
    }
}

// ---------------------------------------------------------------------------
// KNN: 64 blocks = (batch, group of 16 centers), 512 threads = 16 waves.
// V_WMMA_F32_16X16X4_F32 computes full d2 = c2 + p2 - 2*c.p for a 16x16 tile:
//   A row m = (-2cx, -2cy, -2cz, c2) ; B col n = (x, y, z, 1) ; C = p2.
// All 8 subtile point loads are issued before the WMMA stage (latency overlap).
// Wave w then scans center-row w from LDS keeping a wave-distributed sorted
// top-32 (rank r in lane r) with ballot/shfl inserts; ties by (value, index).
// ---------------------------------------------------------------------------
__global__ __launch_bounds__(512) void knn_kernel(const float* __restrict__ points,
                                                  float* __restrict__ out)
{
    extern __shared__ float d2[];          // [16][ROWP] floats = 131,328 B
    __shared__ float sCx[16], sCy[16], sCz[16], sC2[16];

    const int tid  = threadIdx.x, lane = tid & 31, wv = tid >> 5;
    const int b = blockIdx.x >> 2, g = blockIdx.x & 3, c0 = g * 16;
    const float* pts = points + (size_t)b * NPTS * 3;
    const float* ctrAll = out + (size_t)NBATCH * NPATCH * PSIZE * 3;

    if (tid < 16) {
        const float* c = ctrAll + ((size_t)b * NPATCH + c0 + tid) * 3;
        const float x = c[0], y = c[1], z = c[2];
        sCx[tid] = x; sCy[tid] = y; sCz[tid] = z;
        sC2[tid] = fmaf(z, z, fmaf(y, y, x * x));
    }
    __syncthreads();

    const int  m  = lane & 15;
    const bool lo = lane < 16;
    // A-fragment (16x4 f32): lanes 0-15 hold K=0,1; lanes 16-31 hold K=2,3
    v2f A;
    A.x = lo ? (-2.0f * sCx[m]) : (-2.0f * sCz[m]);
    A.y = lo ? (-2.0f * sCy[m]) : sC2[m];

    float bv = FLT_MAX;        // my rank-(lane) value
    int   bi = 0x7fffffff;     // my rank-(lane) point index
    float kv = FLT_MAX;        // rank-31 (threshold), broadcast

    for (int t = 0; t < 32; ++t) {
        const int tb = t * 2048;

        // ---- fill phase: batch all 8 subtile loads, then 8 WMMAs ----
        float xs[8], ys[8], zs[8];
        #pragma unroll
        for (int j = 0; j < 8; ++j) {
            const int gidx = tb + (wv * 8 + j) * 16 + m;
            xs[j] = pts[gidx * 3 + 0];
            ys[j] = pts[gidx * 3 + 1];
            zs[j] = pts[gidx * 3 + 2];
        }
        #pragma unroll
        for (int j = 0; j < 8; ++j) {
            const float x = xs[j], y = ys[j], z = zs[j];
            const float p2 = fmaf(z, z, fmaf(y, y, x * x));
            v2f B;                       // B (4x16): lanes 0-15 K=0,1; 16-31 K=2,3
            B.x = lo ? x : z;
            B.y = lo ? y : 1.0f;
            v8f C = {p2, p2, p2, p2, p2, p2, p2, p2};
            v8f D = __builtin_amdgcn_wmma_f32_16x16x4_f32(
                        false, A, false, B, (short)0, C, false, false);
            // D: VGPR r -> row (r or r+8), column = ln0 + m
            const int rb  = lo ? 0 : 8;
            const int col = (wv * 8 + j) * 16 + m;
            #pragma unroll
            for (int r = 0; r < 8; ++r)
                d2[(rb + r) * ROWP + col] = D[r];
        }
        __syncthreads();

        // ---- selection phase: wave wv scans its center's row ----
        const float4* row4 = (const float4*)(d2 + wv * ROWP);
        #pragma unroll 1
        for (int k4 = 0; k4 < 16; ++k4) {
            const int f4 = k4 * 32 + lane;          // float4 index in row
            const float4 q = row4[f4];
            const int nb = tb + f4 * 4;
            #pragma unroll
            for (int j = 0; j < 4; ++j) {
                const float v  = (j == 0) ? q.x : (j == 1) ? q.y : (j == 2) ? q.z : q.w;
                const int   gi = nb + j;
                unsigned msk = (unsigned)__ballot(v <= kv);
                while (msk) {
                    const int   src = __builtin_ctz(msk);
                    const float cv  = __shfl(v, src);
                    const int   ci  = __shfl(gi, src);
                    const bool less = (bv < cv) || (bv == cv && bi < ci);
                    const int  pos  = __popc((unsigned)__ballot(less));
                    const float pv  = __shfl_up(bv, 1);
                    const int   pi  = __shfl_up(bi, 1);
                    if (lane == pos)     { bv = cv; bi = ci; }
                    else if (lane > pos) { bv = pv; bi = pi; }
                    kv = __shfl(bv, 31);
                    msk &= ~(1u << src);
                    msk &= (unsigned)__ballot(v <= kv);
                }
            }
        }
        __syncthreads();
    }

    // ---- output: lane r holds rank-r neighbor of center c0+wv ----
    const int c = c0 + wv;
    const float cx = sCx[wv], cy = sCy[wv], cz = sCz[wv];
    float* o = out + (((size_t)b * NPATCH + c) * PSIZE + lane) * 3;
    o[0] = pts[bi * 3 + 0] - cx;
    o[1] = pts[bi * 3 + 1] - cy;
    o[2] = pts[bi * 3 + 2] - cz;
}

extern "C" void kernel_launch(void* const* d_in, const int* in_sizes, int n_in,
                              void* d_out, int out_size, void* d_ws, size_t ws_size,
                              hipStream_t stream) {
    const float* points = (const float*)d_in[0];
    float* out = (float*)d_out;
    (void)in_sizes; (void)n_in; (void)out_size; (void)d_ws; (void)ws_size;

    fps_kernel<<<dim3(NBATCH), dim3(1024), NPTS * sizeof(float), stream>>>(points, out);
    knn_kernel<<<dim3(NBATCH * 4), dim3(512), 16 * ROWP * sizeof(float), stream>>>(points, out);
}